// Encoder_Decoder_30580167147770
// MI455X (gfx1250) — compile-verified
//
#include <hip/hip_runtime.h>
#include <hip/hip_bf16.h>

// ---------------------------------------------------------------------------
// Encoder/Decoder GRU pipeline for MI455X (gfx1250, wave32).
//  - Batched GEMMs: V_WMMA_F32_16X16X4_F32 fed by double-buffered
//    GLOBAL_LOAD_ASYNC_TO_LDS_B128 copies (ASYNCcnt-tracked, overlap compute).
//  - Serial GRU recurrences: one persistent workgroup, Whh register-resident,
//    h in LDS, global_prefetch of the next step's precomputed gates.
//  - Attention softmax over [N,1] axis==1 is identically 1.0 => the attended
//    vector is the constant sum of encoder states (never materialize E).
// ---------------------------------------------------------------------------

#define NBOX 4096
#define HID  128

typedef float v2f __attribute__((ext_vector_type(2)));
typedef float v8f __attribute__((ext_vector_type(8)));

__device__ __forceinline__ float sigm_f(float x) {
    return 1.0f / (1.0f + __expf(-x));
}
__device__ __forceinline__ float tanh_f(float x) {
    x = fminf(fmaxf(x, -15.0f), 15.0f);
    float e = __expf(2.0f * x);
    return (e - 1.0f) / (e + 1.0f);
}

// Async global->LDS copy of 16 bytes per lane (CDNA5, ASYNCcnt-tracked).
// vdst VGPR holds the wave-relative LDS byte address (low 32 bits of the
// generic pointer), vaddr holds the 64-bit global address.
__device__ __forceinline__ void async_copy_b128(void* lds, const void* g) {
    asm volatile("global_load_async_to_lds_b128 %0, %1, off"
                 :: "v"((unsigned int)(unsigned long long)(uintptr_t)lds),
                    "v"((unsigned long long)(uintptr_t)g)
                 : "memory");
}
__device__ __forceinline__ void wait_async0() {
#if defined(__has_builtin) && __has_builtin(__builtin_amdgcn_s_wait_asynccnt)
    __builtin_amdgcn_s_wait_asynccnt(0);
#else
    asm volatile("s_wait_asynccnt 0" ::: "memory");
#endif
}

// ---------------------------------------------------------------------------
// fp32 WMMA GEMM:  C[M,N] = act(A[M,K] @ B[K,N] + bias[N])
// Block: 128 threads (4 waves). Block tile 64(M) x 64(N); each wave computes
// a 16x64 strip = 4 WMMA accumulators sharing one A fragment.
// K staged through double-buffered LDS in chunks of 32 via async b128 copies
// (all K here are multiples of 32; all N multiples of 64).
//
// Fragment layouts (CDNA5 ISA 7.12.2, 32-bit, wave32):
//   A 16x4 : lane = {row 0..15} x {khalf}; VGPR0/1 = K = khalf*2 + {0,1}
//   B 4x16 : lane = {col 0..15} x {khalf}; VGPR0/1 = K = khalf*2 + {0,1}
//   C 16x16: VGPR i -> row = i + khalf*8, col = lane & 15
// LDS padding: As stride 36 floats, Bs stride 68 floats -> b128 stores stay
// 16B aligned and fragment reads are bank-conflict-free (mod-64 checked).
// ---------------------------------------------------------------------------
template <int ACT>
__global__ __launch_bounds__(128) void gemm_f32_wmma(
    const float* __restrict__ A, const float* __restrict__ B,
    const float* __restrict__ bias, float* __restrict__ C,
    int M, int N, int K, int lda, int ldb, int ldc)
{
    __shared__ float As[2][64][36];
    __shared__ float Bs[2][32][68];

    const int tid  = threadIdx.x;
    const int wave = tid >> 5;
    const int lane = tid & 31;
    const int half = lane >> 4;     // K-half selector
    const int rc   = lane & 15;     // row (A) / col (B, C)
    const int m0   = blockIdx.y * 64;
    const int n0   = blockIdx.x * 64;
    const int mw   = wave * 16;

    // Staging geometry (one float4 per thread per pass, 4 passes each):
    const int ar = tid >> 3;         // A row base (0..15), +16 per pass
    const int ac = (tid & 7) * 4;    // A col (0..28)
    const int br = tid >> 4;         // B row base (0..7), +8 per pass
    const int bc = (tid & 15) * 4;   // B col (0..60)

    v8f acc0 = {0.f,0.f,0.f,0.f,0.f,0.f,0.f,0.f};
    v8f acc1 = acc0, acc2 = acc0, acc3 = acc0;

    auto stage = [&](int buf, int kk) {
        #pragma unroll
        for (int p = 0; p < 4; ++p) {
            async_copy_b128(&As[buf][ar + 16 * p][ac],
                            A + (m0 + ar + 16 * p) * lda + (kk + ac));
            async_copy_b128(&Bs[buf][br + 8 * p][bc],
                            B + (kk + br + 8 * p) * ldb + (n0 + bc));
        }
    };

    stage(0, 0);
    int buf = 0;
    for (int kk = 0; kk < K; kk += 32) {
        wait_async0();        // this buffer's async copies landed in LDS
        __syncthreads();      // ...and everyone finished reading the other one
        if (kk + 32 < K) stage(buf ^ 1, kk + 32);   // overlap next chunk

        #pragma unroll
        for (int k4 = 0; k4 < 32; k4 += 4) {
            const int ka = k4 + half * 2;
            v2f a, b0, b1, b2, b3;
            a.x  = As[buf][mw + rc][ka];
            a.y  = As[buf][mw + rc][ka + 1];
            b0.x = Bs[buf][ka][rc];          b0.y = Bs[buf][ka + 1][rc];
            b1.x = Bs[buf][ka][16 + rc];     b1.y = Bs[buf][ka + 1][16 + rc];
            b2.x = Bs[buf][ka][32 + rc];     b2.y = Bs[buf][ka + 1][32 + rc];
            b3.x = Bs[buf][ka][48 + rc];     b3.y = Bs[buf][ka + 1][48 + rc];
            acc0 = __builtin_amdgcn_wmma_f32_16x16x4_f32(false, a, false, b0, (short)0, acc0, false, false);
            acc1 = __builtin_amdgcn_wmma_f32_16x16x4_f32(false, a, false, b1, (short)0, acc1, false, false);
            acc2 = __builtin_amdgcn_wmma_f32_16x16x4_f32(false, a, false, b2, (short)0, acc2, false, false);
            acc3 = __builtin_amdgcn_wmma_f32_16x16x4_f32(false, a, false, b3, (short)0, acc3, false, false);
        }
        buf ^= 1;
    }

    v8f accs[4] = {acc0, acc1, acc2, acc3};
    #pragma unroll
    for (int sub = 0; sub < 4; ++sub) {
        const int c  = n0 + sub * 16 + rc;
        const float bv = bias[c];
        #pragma unroll
        for (int i = 0; i < 8; ++i) {
            const int r = m0 + mw + i + half * 8;
            float v = accs[sub][i] + bv;
            if (ACT) v = fmaxf(v, 0.0f);
            C[(size_t)r * ldc + c] = v;
        }
    }
}

// ---------------------------------------------------------------------------
// Persistent sequential GRU scan. One workgroup, 384 threads (12 waves).
// Gi[t] = x_t @ Wih + bih is precomputed (batched WMMA GEMM).
// Thread j keeps column j of Whh (128 floats) in VGPRs; h lives in LDS.
// Encoder mode: accumulates sumE + final hidden. Decoder mode: emits Hout.
// ---------------------------------------------------------------------------
__global__ __launch_bounds__(384) void gru_scan(
    const float* __restrict__ Gi,    // [N, 384] (includes bih)
    const float* __restrict__ Whh,   // [128, 384]
    const float* __restrict__ bhh,   // [384]
    const float* __restrict__ h0,    // [128] or nullptr (zeros)
    float* __restrict__ Hout,        // [N, 128] or nullptr
    float* __restrict__ sumE,        // [128] or nullptr
    float* __restrict__ hfinal,      // [128] or nullptr
    int N)
{
    __shared__ float h[HID];
    __shared__ float gh[3 * HID];
    __shared__ float se[HID];

    const int j = threadIdx.x;   // 0..383

    // Coalesced VGPR preload of Whh column j.
    float w[HID];
    #pragma unroll
    for (int k = 0; k < HID; ++k) w[k] = Whh[k * (3 * HID) + j];
    const float bh = bhh[j];

    if (j < HID) {
        h[j]  = h0 ? h0[j] : 0.0f;
        se[j] = 0.0f;
    }
    __syncthreads();

    for (int t = 0; t < N; ++t) {
        // Pull next step's gates toward the caches while we do the matvec.
        if ((j & 31) == 0 && t + 1 < N)
            __builtin_prefetch(Gi + (size_t)(t + 1) * (3 * HID) + (j >> 5) * 32, 0, 0);

        // gh_j = bhh_j + sum_k h_k * Whh[k][j]   (4-way ILP on the fma chain)
        float a0 = 0.f, a1 = 0.f, a2 = 0.f, a3 = 0.f;
        #pragma unroll
        for (int k = 0; k < HID; k += 4) {
            a0 = fmaf(h[k],     w[k],     a0);
            a1 = fmaf(h[k + 1], w[k + 1], a1);
            a2 = fmaf(h[k + 2], w[k + 2], a2);
            a3 = fmaf(h[k + 3], w[k + 3], a3);
        }
        gh[j] = bh + ((a0 + a1) + (a2 + a3));
        __syncthreads();   // gh ready; all h reads for this step retired

        if (j < HID) {
            const float* gi = Gi + (size_t)t * (3 * HID);
            float r  = sigm_f(gi[j]           + gh[j]);
            float z  = sigm_f(gi[HID + j]     + gh[HID + j]);
            float nn = tanh_f(gi[2 * HID + j] + r * gh[2 * HID + j]);
            float hn = (1.0f - z) * nn + z * h[j];
            h[j] = hn;
            se[j] += hn;
            if (Hout) Hout[(size_t)t * HID + j] = hn;
        }
        __syncthreads();   // h updated before next step's matvec
    }

    if (j < HID) {
        if (sumE)   sumE[j]   = se[j];
        if (hfinal) hfinal[j] = h[j];
    }
}

// c2[j] = b_comb[j] + sum_k sumE[k] * W_comb[128 + k][j]   (constant attended part)
__global__ __launch_bounds__(128) void comb_const_kernel(
    const float* __restrict__ sumE, const float* __restrict__ W_comb,
    const float* __restrict__ b_comb, float* __restrict__ c2)
{
    const int j = threadIdx.x;
    float acc = b_comb[j];
    #pragma unroll 8
    for (int k = 0; k < HID; ++k)
        acc = fmaf(sumE[k], W_comb[(HID + k) * HID + j], acc);
    c2[j] = acc;
}

// out[t] = sigmoid(D[t] . W_out + b_out); pass-through label/weight.
__global__ __launch_bounds__(256) void out_head_kernel(
    const float* __restrict__ D, const float* __restrict__ W_out,
    const float* __restrict__ b_out, const float* __restrict__ label,
    const float* __restrict__ weight, float* __restrict__ out, int N)
{
    const int t = blockIdx.x * blockDim.x + threadIdx.x;
    if (t >= N) return;
    float acc = 0.0f;
    #pragma unroll 8
    for (int jj = 0; jj < HID; ++jj)
        acc = fmaf(D[t * HID + jj], W_out[jj], acc);
    out[t]         = sigm_f(acc + b_out[0]);
    out[N + t]     = label[t];
    out[2 * N + t] = weight[t];
}

// ---------------------------------------------------------------------------

static inline void launch_gemm(int act, const float* A, const float* B,
                               const float* bias, float* C,
                               int M, int N, int K, int lda, int ldb, int ldc,
                               hipStream_t s)
{
    dim3 grid((unsigned)(N / 64), (unsigned)(M / 64)), blk(128);
    if (act)
        hipLaunchKernelGGL((gemm_f32_wmma<1>), grid, blk, 0, s,
                           A, B, bias, C, M, N, K, lda, ldb, ldc);
    else
        hipLaunchKernelGGL((gemm_f32_wmma<0>), grid, blk, 0, s,
                           A, B, bias, C, M, N, K, lda, ldb, ldc);
}

extern "C" void kernel_launch(void* const* d_in, const int* in_sizes, int n_in,
                              void* d_out, int out_size, void* d_ws, size_t ws_size,
                              hipStream_t stream)
{
    const float* boxes_feature   = (const float*)d_in[0];   // [1,4096,1024]
    const float* boxes_box_score = (const float*)d_in[1];   // [1,4096,2560]
    const float* boxes_box       = (const float*)d_in[2];   // [1,4096,320]
    const float* boxes_label     = (const float*)d_in[3];   // [1,4096]
    const float* boxes_weight    = (const float*)d_in[4];   // [1,4096]
    // d_in[5] = boxes_num (int, == 4096, compile-time constant here)
    const float* W_appear = (const float*)d_in[6];
    const float* b_appear = (const float*)d_in[7];
    const float* W_s1     = (const float*)d_in[8];
    const float* b_s1     = (const float*)d_in[9];
    const float* W_s2     = (const float*)d_in[10];
    const float* b_s2     = (const float*)d_in[11];
    const float* W_box    = (const float*)d_in[12];
    const float* b_box    = (const float*)d_in[13];
    const float* W_all    = (const float*)d_in[14];
    const float* b_all    = (const float*)d_in[15];
    const float* enc_Wih  = (const float*)d_in[16];
    const float* enc_Whh  = (const float*)d_in[17];
    const float* enc_bih  = (const float*)d_in[18];
    const float* enc_bhh  = (const float*)d_in[19];
    const float* dec_Wih  = (const float*)d_in[20];
    const float* dec_Whh  = (const float*)d_in[21];
    const float* dec_bih  = (const float*)d_in[22];
    const float* dec_bhh  = (const float*)d_in[23];
    // d_in[24]/d_in[25]: W_attn/b_attn — unused: softmax over axis=1 of [N,1]
    // scores is identically 1.0, so attention weights never affect the output.
    const float* W_comb   = (const float*)d_in[26];
    const float* b_comb   = (const float*)d_in[27];
    const float* W_out    = (const float*)d_in[28];
    const float* b_out    = (const float*)d_in[29];

    float* out = (float*)d_out;
    float* ws  = (float*)d_ws;

    // Workspace layout (floats): ~27.3 MB total
    float* gs   = ws;                        // [4096, 512] score MLP hidden
    float* Cat  = gs   + (size_t)NBOX * 512; // [4096, 384] concat(ef, es, eb)
    float* X    = Cat  + (size_t)NBOX * 384; // [4096, 128]
    float* Gi   = X    + (size_t)NBOX * 128; // [4096, 384] (reused enc then dec)
    float* Din  = Gi   + (size_t)NBOX * 384; // [4096, 128]
    float* Dmat = Din  + (size_t)NBOX * 128; // [4096, 128] decoder states
    float* sumE = Dmat + (size_t)NBOX * 128; // [128]
    float* hEnc = sumE + HID;                // [128]
    float* c2   = hEnc + HID;                // [128]

    // 1) gs = score @ W_s1 + b_s1            (4096x2560 x 2560x512, no act)
    launch_gemm(0, boxes_box_score, W_s1, b_s1, gs,
                NBOX, 512, 2560, 2560, 512, 512, stream);
    // 2) ef = relu(feat @ W_appear + b)  -> Cat[:, 0:128]
    launch_gemm(1, boxes_feature, W_appear, b_appear, Cat + 0,
                NBOX, 128, 1024, 1024, 128, 384, stream);
    // 3) es = relu(gs @ W_s2 + b)        -> Cat[:, 128:256]
    launch_gemm(1, gs, W_s2, b_s2, Cat + 128,
                NBOX, 128, 512, 512, 128, 384, stream);
    // 4) eb = relu(box @ W_box + b)      -> Cat[:, 256:384]
    launch_gemm(1, boxes_box, W_box, b_box, Cat + 256,
                NBOX, 128, 320, 320, 128, 384, stream);
    // 5) X = relu(Cat @ W_all + b_all)
    launch_gemm(1, Cat, W_all, b_all, X,
                NBOX, 128, 384, 384, 128, 128, stream);
    // 6) Gi = X @ enc_Wih + enc_bih   (batched input half of encoder GRU)
    launch_gemm(0, X, enc_Wih, enc_bih, Gi,
                NBOX, 384, 128, 128, 384, 384, stream);
    // 7) Encoder scan: sumE (= attention "applied", weights all 1) + h_enc
    hipLaunchKernelGGL(gru_scan, dim3(1), dim3(384), 0, stream,
                       Gi, enc_Whh, enc_bhh, (const float*)nullptr,
                       (float*)nullptr, sumE, hEnc, NBOX);
    // 8) c2 = sumE @ W_comb[128:] + b_comb (constant over decoder steps)
    hipLaunchKernelGGL(comb_const_kernel, dim3(1), dim3(128), 0, stream,
                       sumE, W_comb, b_comb, c2);
    // 9) Din = relu(X @ W_comb[:128] + c2)
    launch_gemm(1, X, W_comb, c2, Din,
                NBOX, 128, 128, 128, 128, 128, stream);
    // 10) Gi = Din @ dec_Wih + dec_bih
    launch_gemm(0, Din, dec_Wih, dec_bih, Gi,
                NBOX, 384, 128, 128, 384, 384, stream);
    // 11) Decoder scan from h_enc -> D
    hipLaunchKernelGGL(gru_scan, dim3(1), dim3(384), 0, stream,
                       Gi, dec_Whh, dec_bhh, hEnc,
                       Dmat, (float*)nullptr, (float*)nullptr, NBOX);
    // 12) out = sigmoid(D @ W_out + b_out); append label, weight
    hipLaunchKernelGGL(out_head_kernel, dim3(NBOX / 256), dim3(256), 0, stream,
                       Dmat, W_out, b_out, boxes_label, boxes_weight, out, NBOX);
}